// GCN4line_graph_61306363183623
// MI455X (gfx1250) — compile-verified
//
#include <hip/hip_runtime.h>
#include <hip/hip_bf16.h>

typedef __attribute__((ext_vector_type(2))) float v2f;
typedef __attribute__((ext_vector_type(8))) float v8f;

// ---------------------------------------------------------------------------
// FP32 WMMA GEMM: H[N,M] = X[N,K] @ W[K,M]. M must be a multiple of 16.
// One wave computes one 16x16 output tile via V_WMMA_F32_16X16X4_F32.
// blockDim.x = 32 * (M/16); blockIdx.x = row tile.
// A layout (16x4 f32): lanes 0-15 row M=lane, K offsets {0,1}; lanes 16-31
// row M=lane-16, K offsets {2,3}. B mirrored. C/D: VGPR r holds row r (lanes
// 0-15) and row r+8 (lanes 16-31), N = lane&15.
// Main K loop is branch-free; odd-K tail uses clamped-address loads + select
// so EXEC stays all-1s (WMMA requirement) and no exec-mask ladders appear.
// ---------------------------------------------------------------------------
__global__ void gemm_wmma_f32(const float* __restrict__ X,
                              const float* __restrict__ W,
                              float* __restrict__ H,
                              int N, int K, int M) {
    const int lane    = threadIdx.x & 31;
    const int wave    = threadIdx.x >> 5;
    const int rowBase = blockIdx.x * 16;
    const int colBase = wave * 16;
    if (rowBase >= N) return;            // uniform per block -> EXEC stays full

    const int l15 = lane & 15;
    const int kh  = (lane >> 4) << 1;    // lanes 0-15 -> K offset 0,1 ; lanes 16-31 -> 2,3
    const int col = colBase + l15;       // always < M

    int rsrc = rowBase + l15;            // clamp tail rows (reads only; stores guarded)
    if (rsrc >= N) rsrc = N - 1;
    const float* xrow = X + (size_t)rsrc * K;

    v8f c = {0.f, 0.f, 0.f, 0.f, 0.f, 0.f, 0.f, 0.f};

    const int K4 = K & ~3;
    for (int k = 0; k < K4; k += 4) {
        const int ka = k + kh;
        v2f a, b;
        a.x = xrow[ka];
        a.y = xrow[ka + 1];
        b.x = W[(size_t)ka * M + col];
        b.y = W[(size_t)(ka + 1) * M + col];
        c = __builtin_amdgcn_wmma_f32_16x16x4_f32(false, a, false, b,
                                                  (short)0, c, false, false);
    }
    if (K4 < K) {                        // odd-K tail: clamped loads + select
        const int ka = K4 + kh;
        const int kb = ka + 1;
        const int kaC = (ka < K) ? ka : (K - 1);
        const int kbC = (kb < K) ? kb : (K - 1);
        const float ax = xrow[kaC];
        const float ay = xrow[kbC];
        const float bx = W[(size_t)kaC * M + col];
        const float by = W[(size_t)kbC * M + col];
        v2f a, b;
        a.x = (ka < K) ? ax : 0.f;
        a.y = (kb < K) ? ay : 0.f;
        b.x = (ka < K) ? bx : 0.f;
        b.y = (kb < K) ? by : 0.f;
        c = __builtin_amdgcn_wmma_f32_16x16x4_f32(false, a, false, b,
                                                  (short)0, c, false, false);
    }

    const int rowHalf = (lane >> 4) << 3;   // 0 or 8
#pragma unroll
    for (int r = 0; r < 8; ++r) {
        const int row = rowBase + r + rowHalf;
        if (row < N) H[(size_t)row * M + col] = c[r];
    }
}

// ---------------------------------------------------------------------------
// Narrow GEMM for M == 5 (64->5): one thread per row, W broadcast via cache.
// ---------------------------------------------------------------------------
__global__ void gemm_n5(const float* __restrict__ X, const float* __restrict__ W,
                        float* __restrict__ H, int N, int K) {
    const int i = blockIdx.x * blockDim.x + threadIdx.x;
    if (i >= N) return;
    const float* xr = X + (size_t)i * K;
    float a0 = 0.f, a1 = 0.f, a2 = 0.f, a3 = 0.f, a4 = 0.f;
    for (int k = 0; k < K; ++k) {
        const float xv = xr[k];
        const float* wr = W + (size_t)k * 5;
        a0 += xv * wr[0]; a1 += xv * wr[1]; a2 += xv * wr[2];
        a3 += xv * wr[3]; a4 += xv * wr[4];
    }
    float* o = H + (size_t)i * 5;
    o[0] = a0; o[1] = a1; o[2] = a2; o[3] = a3; o[4] = a4;
}

// ---------------------------------------------------------------------------
// Degree / normalization
// ---------------------------------------------------------------------------
__global__ void zero_f32(float* __restrict__ p, long long n) {
    long long i = (long long)blockIdx.x * blockDim.x + threadIdx.x;
    if (i < n) p[i] = 0.f;
}

__global__ void deg_count(const long long* __restrict__ dst, float* __restrict__ deg,
                          long long E) {
    long long e = (long long)blockIdx.x * blockDim.x + threadIdx.x;
    if (e < E) atomicAdd(&deg[(int)dst[e]], 1.0f);
}

__global__ void finish_dinv(float* __restrict__ deg, int N) {
    int i = blockIdx.x * blockDim.x + threadIdx.x;
    if (i < N) deg[i] = rsqrtf(deg[i] + 1.0f);   // +1 for self loop; always > 0
}

// ---------------------------------------------------------------------------
// Aggregation: init with self-loop message, scatter edges, then bias(+ReLU).
// ---------------------------------------------------------------------------
__global__ void agg_init(const float* __restrict__ h, const float* __restrict__ dinv,
                         float* __restrict__ out, int N, int F) {
    long long idx = (long long)blockIdx.x * blockDim.x + threadIdx.x;
    if (idx >= (long long)N * F) return;
    const int i = (int)(idx / F);
    const float d = dinv[i];
    out[idx] = h[idx] * d * d;
}

// F divisible by 4; qshift = log2(F/4). 32-bit node-index math (N = 100k).
__global__ void scatter_vec4(const float4* __restrict__ h4,
                             const long long* __restrict__ src,
                             const long long* __restrict__ dst,
                             const float* __restrict__ dinv,
                             float* __restrict__ out,
                             long long E, int qshift) {
    long long idx = (long long)blockIdx.x * blockDim.x + threadIdx.x;
    const long long total = E << qshift;
    if (idx >= total) return;
    const long long e = idx >> qshift;
    const int q = (int)(idx & ((1 << qshift) - 1));
    const int s = (int)src[e];
    const int d = (int)dst[e];
    const float nrm = dinv[s] * dinv[d];
    const float4 v = h4[(unsigned)((s << qshift) + q)];
    float* o = out + ((size_t)(unsigned)((d << qshift) + q) << 2);
    atomicAdd(o + 0, v.x * nrm);
    atomicAdd(o + 1, v.y * nrm);
    atomicAdd(o + 2, v.z * nrm);
    atomicAdd(o + 3, v.w * nrm);
}

__global__ void scatter_scalar(const float* __restrict__ h,
                               const long long* __restrict__ src,
                               const long long* __restrict__ dst,
                               const float* __restrict__ dinv,
                               float* __restrict__ out,
                               long long E, int F) {
    long long idx = (long long)blockIdx.x * blockDim.x + threadIdx.x;
    if (idx >= E * (long long)F) return;
    const long long e = idx / F;
    const int f = (int)(idx % F);
    const int s = (int)src[e];
    const int d = (int)dst[e];
    atomicAdd(&out[(unsigned)(d * F + f)], h[(unsigned)(s * F + f)] * dinv[s] * dinv[d]);
}

__global__ void bias_act(float* __restrict__ out, const float* __restrict__ b,
                         int N, int F, int relu) {
    long long idx = (long long)blockIdx.x * blockDim.x + threadIdx.x;
    if (idx >= (long long)N * F) return;
    const int f = (int)(idx % F);
    float v = out[idx] + b[f];
    if (relu) v = fmaxf(v, 0.f);
    out[idx] = v;
}

// ---------------------------------------------------------------------------
// Global mean pool (F == 5) into gsum[64][10] at column offset colOff.
// ---------------------------------------------------------------------------
__global__ void pool_scatter(const float* __restrict__ h,
                             const long long* __restrict__ batch,
                             float* __restrict__ gsum, float* __restrict__ cnt,
                             int N, int colOff) {
    const int i = blockIdx.x * blockDim.x + threadIdx.x;
    if (i >= N) return;
    const int g = (int)batch[i];
    atomicAdd(&cnt[g], 1.0f);
    const float* hr = h + (size_t)i * 5;
#pragma unroll
    for (int f = 0; f < 5; ++f) atomicAdd(&gsum[g * 10 + colOff + f], hr[f]);
}

// ---------------------------------------------------------------------------
// Head: g[64,10] @ Wfc[10,2] + bfc -> log_softmax -> out[64,2]
// ---------------------------------------------------------------------------
__global__ void head_kernel(const float* __restrict__ gsum,
                            const float* __restrict__ cnt_o,
                            const float* __restrict__ cnt_l,
                            const float* __restrict__ Wfc,
                            const float* __restrict__ bfc,
                            float* __restrict__ out, int G) {
    const int g = blockIdx.x * blockDim.x + threadIdx.x;
    if (g >= G) return;
    const float co = fmaxf(cnt_o[g], 1.0f);
    const float cl = fmaxf(cnt_l[g], 1.0f);
    float z0 = bfc[0], z1 = bfc[1];
#pragma unroll
    for (int f = 0; f < 10; ++f) {
        const float feat = gsum[g * 10 + f] / ((f < 5) ? co : cl);
        z0 += feat * Wfc[f * 2 + 0];
        z1 += feat * Wfc[f * 2 + 1];
    }
    const float m = fmaxf(z0, z1);
    const float lse = m + logf(expf(z0 - m) + expf(z1 - m));
    out[g * 2 + 0] = z0 - lse;
    out[g * 2 + 1] = z1 - lse;
}

// ---------------------------------------------------------------------------
static inline unsigned int blocks_for(long long total, int bs) {
    return (unsigned int)((total + bs - 1) / bs);
}

extern "C" void kernel_launch(void* const* d_in, const int* in_sizes, int n_in,
                              void* d_out, int out_size, void* d_ws, size_t ws_size,
                              hipStream_t stream) {
    const float*     x_o     = (const float*)d_in[0];
    const long long* ei_o    = (const long long*)d_in[1];
    const long long* batch_o = (const long long*)d_in[2];
    const float*     x_l     = (const float*)d_in[3];
    const long long* ei_l    = (const long long*)d_in[4];
    const long long* batch_l = (const long long*)d_in[5];
    const float* W1 = (const float*)d_in[6];   const float* b1 = (const float*)d_in[7];
    const float* W2 = (const float*)d_in[8];   const float* b2 = (const float*)d_in[9];
    const float* W5 = (const float*)d_in[10];  const float* b5 = (const float*)d_in[11];
    const float* W3 = (const float*)d_in[12];  const float* b3 = (const float*)d_in[13];
    const float* W4 = (const float*)d_in[14];  const float* b4 = (const float*)d_in[15];
    const float* Wfc = (const float*)d_in[16]; const float* bfc = (const float*)d_in[17];

    const int       N = in_sizes[2];                 // 100000 nodes
    const long long E = (long long)in_sizes[1] / 2;  // 1.6M edges
    const int       G = 64;                          // NUM_GRAPHS
    const long long* src_o = ei_o;     const long long* dst_o = ei_o + E;
    const long long* src_l = ei_l;     const long long* dst_l = ei_l + E;

    // Workspace carve-up (≈103 MB): two N*128 ping-pong buffers + dinv + pool.
    char* ws = (char*)d_ws;
    const size_t nb = (size_t)N * 128 * sizeof(float);
    float* bufA   = (float*)(ws);
    float* bufB   = (float*)(ws + nb);
    float* dinv_o = (float*)(ws + 2 * nb);
    float* dinv_l = dinv_o + N;
    float* gsum   = dinv_l + N;          // 64*10
    float* cnt_o  = gsum + G * 10;       // 64
    float* cnt_l  = cnt_o + G;           // 64  (gsum/cnt contiguous: 768 floats)

    const int BS = 256;

    // --- degrees -> dinv (shared across all convs of a branch) --------------
    zero_f32<<<blocks_for(N, BS), BS, 0, stream>>>(dinv_o, N);
    zero_f32<<<blocks_for(N, BS), BS, 0, stream>>>(dinv_l, N);
    zero_f32<<<blocks_for(G * 10 + 2 * G, BS), BS, 0, stream>>>(gsum, G * 10 + 2 * G);
    deg_count<<<blocks_for(E, BS), BS, 0, stream>>>(dst_o, dinv_o, E);
    deg_count<<<blocks_for(E, BS), BS, 0, stream>>>(dst_l, dinv_l, E);
    finish_dinv<<<blocks_for(N, BS), BS, 0, stream>>>(dinv_o, N);
    finish_dinv<<<blocks_for(N, BS), BS, 0, stream>>>(dinv_l, N);

    const unsigned int rowTiles = blocks_for(N, 16);

    // ======================= origin branch ==================================
    // conv1: 25 -> 128, ReLU
    gemm_wmma_f32<<<rowTiles, 32 * (128 / 16), 0, stream>>>(x_o, W1, bufA, N, 25, 128);
    agg_init<<<blocks_for((long long)N * 128, BS), BS, 0, stream>>>(bufA, dinv_o, bufB, N, 128);
    scatter_vec4<<<blocks_for(E << 5, BS), BS, 0, stream>>>((const float4*)bufA, src_o, dst_o,
                                                            dinv_o, bufB, E, 5);
    bias_act<<<blocks_for((long long)N * 128, BS), BS, 0, stream>>>(bufB, b1, N, 128, 1);

    // conv2: 128 -> 64, ReLU
    gemm_wmma_f32<<<rowTiles, 32 * (64 / 16), 0, stream>>>(bufB, W2, bufA, N, 128, 64);
    agg_init<<<blocks_for((long long)N * 64, BS), BS, 0, stream>>>(bufA, dinv_o, bufB, N, 64);
    scatter_vec4<<<blocks_for(E << 4, BS), BS, 0, stream>>>((const float4*)bufA, src_o, dst_o,
                                                            dinv_o, bufB, E, 4);
    bias_act<<<blocks_for((long long)N * 64, BS), BS, 0, stream>>>(bufB, b2, N, 64, 1);

    // conv5: 64 -> 5, no activation
    gemm_n5<<<blocks_for(N, BS), BS, 0, stream>>>(bufB, W5, bufA, N, 64);
    agg_init<<<blocks_for((long long)N * 5, BS), BS, 0, stream>>>(bufA, dinv_o, bufB, N, 5);
    scatter_scalar<<<blocks_for(E * 5, BS), BS, 0, stream>>>(bufA, src_o, dst_o, dinv_o,
                                                             bufB, E, 5);
    bias_act<<<blocks_for((long long)N * 5, BS), BS, 0, stream>>>(bufB, b5, N, 5, 0);
    pool_scatter<<<blocks_for(N, BS), BS, 0, stream>>>(bufB, batch_o, gsum, cnt_o, N, 0);

    // ======================== line branch ===================================
    // conv3: 51 -> 64, ReLU
    gemm_wmma_f32<<<rowTiles, 32 * (64 / 16), 0, stream>>>(x_l, W3, bufA, N, 51, 64);
    agg_init<<<blocks_for((long long)N * 64, BS), BS, 0, stream>>>(bufA, dinv_l, bufB, N, 64);
    scatter_vec4<<<blocks_for(E << 4, BS), BS, 0, stream>>>((const float4*)bufA, src_l, dst_l,
                                                            dinv_l, bufB, E, 4);
    bias_act<<<blocks_for((long long)N * 64, BS), BS, 0, stream>>>(bufB, b3, N, 64, 1);

    // conv4: 64 -> 5, no activation
    gemm_n5<<<blocks_for(N, BS), BS, 0, stream>>>(bufB, W4, bufA, N, 64);
    agg_init<<<blocks_for((long long)N * 5, BS), BS, 0, stream>>>(bufA, dinv_l, bufB, N, 5);
    scatter_scalar<<<blocks_for(E * 5, BS), BS, 0, stream>>>(bufA, src_l, dst_l, dinv_l,
                                                             bufB, E, 5);
    bias_act<<<blocks_for((long long)N * 5, BS), BS, 0, stream>>>(bufB, b4, N, 5, 0);
    pool_scatter<<<blocks_for(N, BS), BS, 0, stream>>>(bufB, batch_l, gsum, cnt_l, N, 5);

    // ============================ head ======================================
    head_kernel<<<1, G, 0, stream>>>(gsum, cnt_o, cnt_l, Wfc, bfc, (float*)d_out, G);
}